// GATON_37074157699558
// MI455X (gfx1250) — compile-verified
//
#include <hip/hip_runtime.h>
#include <hip/hip_bf16.h>
#include <math.h>

typedef __attribute__((ext_vector_type(2))) float v2f;
typedef __attribute__((ext_vector_type(4))) float v4f;
typedef __attribute__((ext_vector_type(8))) float v8f;

// ---------------------------------------------------------------------------
// fp32 WMMA GEMM:  C[M,N] = A[M,K] @ W[N,K]^T (+ bias), one wave per 16x64 tile
// A-tile 16x4 / B-tile 4x16 per V_WMMA_F32_16X16X4_F32 VGPR layout:
//   lanes 0-15:  row (m0|n0)+lane, K pair {k,k+1}; lanes 16-31: K pair {k+2,k+3}
// Software-pipelined: loads for step k+4 are issued before the 4 WMMAs of
// step k, so s_wait_loadcnt overlaps matrix math instead of preceding it.
// ---------------------------------------------------------------------------
__global__ void gemm16x64_wmma(const float* __restrict__ A, const float* __restrict__ W,
                               const float* __restrict__ bias, float* __restrict__ C,
                               int M, int N, int K) {
  const int lane = threadIdx.x & 31;
  const int wave = blockIdx.x * (blockDim.x >> 5) + (threadIdx.x >> 5);
  const int ntg  = N >> 6;                       // 64-wide column groups
  const int mt   = wave / ntg;
  const int ng   = wave - mt * ntg;
  if (mt >= (M >> 4)) return;                    // wave-uniform guard (EXEC stays all-1)
  const int m0 = mt << 4, n0 = ng << 6;
  const int l15   = lane & 15;
  const int khalf = (lane >> 4) << 1;            // 0 for lanes 0-15, 2 for 16-31
  const float* Ap  = A + (size_t)(m0 + l15) * K + khalf;
  const float* Wp0 = W + (size_t)(n0 + l15) * K + khalf;
  const float* Wp1 = Wp0 + (size_t)16 * K;
  const float* Wp2 = Wp0 + (size_t)32 * K;
  const float* Wp3 = Wp0 + (size_t)48 * K;
  v8f acc0 = {}, acc1 = {}, acc2 = {}, acc3 = {};
  v2f a  = *(const v2f*)(Ap);
  v2f b0 = *(const v2f*)(Wp0);
  v2f b1 = *(const v2f*)(Wp1);
  v2f b2 = *(const v2f*)(Wp2);
  v2f b3 = *(const v2f*)(Wp3);
  int k = 0;
  for (; k + 4 < K; k += 4) {
    const int kn = k + 4;
    __builtin_prefetch(Ap + kn + 252, 0, 0);     // global_prefetch_b8, ~1KB ahead
    v2f an  = *(const v2f*)(Ap  + kn);
    v2f bn0 = *(const v2f*)(Wp0 + kn);
    v2f bn1 = *(const v2f*)(Wp1 + kn);
    v2f bn2 = *(const v2f*)(Wp2 + kn);
    v2f bn3 = *(const v2f*)(Wp3 + kn);
    acc0 = __builtin_amdgcn_wmma_f32_16x16x4_f32(false, a, false, b0, (short)0, acc0, false, false);
    acc1 = __builtin_amdgcn_wmma_f32_16x16x4_f32(false, a, false, b1, (short)0, acc1, false, false);
    acc2 = __builtin_amdgcn_wmma_f32_16x16x4_f32(false, a, false, b2, (short)0, acc2, false, false);
    acc3 = __builtin_amdgcn_wmma_f32_16x16x4_f32(false, a, false, b3, (short)0, acc3, false, false);
    a = an; b0 = bn0; b1 = bn1; b2 = bn2; b3 = bn3;
  }
  // epilogue step (k == K-4)
  acc0 = __builtin_amdgcn_wmma_f32_16x16x4_f32(false, a, false, b0, (short)0, acc0, false, false);
  acc1 = __builtin_amdgcn_wmma_f32_16x16x4_f32(false, a, false, b1, (short)0, acc1, false, false);
  acc2 = __builtin_amdgcn_wmma_f32_16x16x4_f32(false, a, false, b2, (short)0, acc2, false, false);
  acc3 = __builtin_amdgcn_wmma_f32_16x16x4_f32(false, a, false, b3, (short)0, acc3, false, false);

  const int rbase = m0 + ((lane >> 4) << 3);     // C/D layout: VGPR r -> row r (+8 for hi lanes)
#pragma unroll
  for (int t = 0; t < 4; ++t) {
    const v8f& acc = (t == 0) ? acc0 : (t == 1) ? acc1 : (t == 2) ? acc2 : acc3;
    const int col = n0 + t * 16 + l15;
    const float bv = bias ? bias[col] : 0.0f;
#pragma unroll
    for (int r = 0; r < 8; ++r)
      C[(size_t)(rbase + r) * N + col] = acc[r] + bv;
  }
}

// ---------------------------------------------------------------------------
// attention score:  out[n,h] = sum_c H[n, h*C+c] * att[h,c]   (one wave per (n,h))
// ---------------------------------------------------------------------------
__global__ void attn_score(const float* __restrict__ Hm, const float* __restrict__ att,
                           float* __restrict__ out, int Nrows, int Hh, int Cc) {
  const int wid  = blockIdx.x * (blockDim.x >> 5) + (threadIdx.x >> 5);
  const int lane = threadIdx.x & 31;
  if (wid >= Nrows * Hh) return;
  const int n = wid / Hh, h = wid - n * Hh;
  const float* row = Hm + (size_t)n * (Hh * Cc) + (size_t)h * Cc;
  const float* a   = att + (size_t)h * Cc;
  float s = 0.0f;
  for (int c = lane; c < Cc; c += 32) s += row[c] * a[c];
  for (int o = 16; o; o >>= 1) s += __shfl_xor(s, o, 32);
  if (lane == 0) out[wid] = s;
}

__global__ void fill_f32(float* __restrict__ p, float v, int n) {
  int i = blockIdx.x * blockDim.x + threadIdx.x;
  if (i < n) p[i] = v;
}

__device__ __forceinline__ void atomicMaxF(float* addr, float val) {
  if (val >= 0.0f) atomicMax((int*)addr, __float_as_int(val));
  else             atomicMin((unsigned int*)addr, __float_as_uint(val));
}

// alpha = leaky_relu(a_s[src]+a_d[dst], 0.2); segment-max into amax[dst]
__global__ void edge_alpha_max(const int* __restrict__ es, const int* __restrict__ ed,
                               const float* __restrict__ AS, const float* __restrict__ AD,
                               float* __restrict__ alpha, float* __restrict__ amax,
                               int E, int nself, int Hh) {
  int idx = blockIdx.x * blockDim.x + threadIdx.x;
  int tot = (E + nself) * Hh;
  if (idx >= tot) return;
  int e = idx / Hh, h = idx - e * Hh;
  int s = (e < E) ? es[e] : (e - E);
  int d = (e < E) ? ed[e] : (e - E);
  float x = AS[s * Hh + h] + AD[d * Hh + h];
  x = (x > 0.0f) ? x : 0.2f * x;
  alpha[idx] = x;
  atomicMaxF(&amax[d * Hh + h], x);
}

// ex = exp(alpha - amax[dst]); segment-sum into denom[dst]; alpha := ex
__global__ void edge_exp_sum(const int* __restrict__ ed, float* __restrict__ alpha,
                             const float* __restrict__ amax, float* __restrict__ denom,
                             int E, int nself, int Hh) {
  int idx = blockIdx.x * blockDim.x + threadIdx.x;
  int tot = (E + nself) * Hh;
  if (idx >= tot) return;
  int e = idx / Hh, h = idx - e * Hh;
  int d = (e < E) ? ed[e] : (e - E);
  float ex = __expf(alpha[idx] - amax[d * Hh + h]);
  alpha[idx] = ex;
  atomicAdd(&denom[d * Hh + h], ex);
}

// out[dst, j..j+3] += hs[src, j..j+3] * ex/(denom[dst]+eps)
// thread = (edge, 4-channel group): float4 gather (global_load_b128) +
// 4 consecutive global_atomic_add_f32 (L2 coalesces the 16B span)
__global__ void edge_scatter4(const int* __restrict__ es, const int* __restrict__ ed,
                              const float* __restrict__ ex, const float* __restrict__ denom,
                              const float* __restrict__ HS, float* __restrict__ out,
                              int E, int nself, int Hh, int Cc) {
  size_t idx = (size_t)blockIdx.x * blockDim.x + threadIdx.x;
  const int HC  = Hh * Cc;
  const int HC4 = HC >> 2;
  size_t tot = (size_t)(E + nself) * HC4;
  if (idx >= tot) return;
  int e  = (int)(idx / HC4);
  int j4 = (int)(idx - (size_t)e * HC4);
  int j  = j4 << 2;
  int h  = j / Cc;
  int s = (e < E) ? es[e] : (e - E);
  int d = (e < E) ? ed[e] : (e - E);
  float w = ex[(size_t)e * Hh + h] / (denom[d * Hh + h] + 1e-16f);
  const v4f hv = *(const v4f*)(HS + (size_t)s * HC + j);
  float* o = out + (size_t)d * HC + j;
  atomicAdd(o + 0, hv[0] * w);
  atomicAdd(o + 1, hv[1] * w);
  atomicAdd(o + 2, hv[2] * w);
  atomicAdd(o + 3, hv[3] * w);
}

__global__ void bias_act_elu(float* __restrict__ out, const float* __restrict__ bias,
                             int Nrows, int HC) {
  size_t idx = (size_t)blockIdx.x * blockDim.x + threadIdx.x;
  if (idx >= (size_t)Nrows * HC) return;
  int j = (int)(idx % HC);
  float x = out[idx] + bias[j];
  out[idx] = (x > 0.0f) ? x : (__expf(x) - 1.0f);
}

// row softmax over 64 channels, one wave per row, 2 channels/lane, in place
__global__ void bias_softmax64(float* __restrict__ out, const float* __restrict__ bias, int Nrows) {
  const int wid  = blockIdx.x * (blockDim.x >> 5) + (threadIdx.x >> 5);
  const int lane = threadIdx.x & 31;
  if (wid >= Nrows) return;
  float* row = out + (size_t)wid * 64;
  float v0 = row[lane]      + bias[lane];
  float v1 = row[lane + 32] + bias[lane + 32];
  float m = fmaxf(v0, v1);
  for (int o = 16; o; o >>= 1) m = fmaxf(m, __shfl_xor(m, o, 32));
  float e0 = __expf(v0 - m), e1 = __expf(v1 - m);
  float s = e0 + e1;
  for (int o = 16; o; o >>= 1) s += __shfl_xor(s, o, 32);
  row[lane] = e0 / s;
  row[lane + 32] = e1 / s;
}

// ---------------------------------------------------------------------------
// host-side orchestration
// ---------------------------------------------------------------------------
static inline int cdivl(long a, long b) { return (int)((a + b - 1) / b); }

static void launch_gemm(hipStream_t s, const float* A, const float* W, const float* bias,
                        float* C, int M, int N, int K) {
  long waves = (long)(M >> 4) * (N >> 6);
  gemm16x64_wmma<<<cdivl(waves, 4), 128, 0, s>>>(A, W, bias, C, M, N, K);
}

struct ConvScratch { float *HS, *HD, *AS, *AD, *AMAX, *DENOM, *ALPHA; };

static void run_conv(hipStream_t s,
                     const float* x_src, int n_src, const float* x_dst, int n_dst, int Kin,
                     const float* w_src, const float* w_dst,
                     const float* att_src, const float* att_dst,
                     const int* es, const int* ed, int E, int nself,
                     int H, int C, float* OUT, const ConvScratch& w) {
  const int HC = H * C;
  launch_gemm(s, x_src, w_src, nullptr, w.HS, n_src, HC, Kin);
  launch_gemm(s, x_dst, w_dst, nullptr, w.HD, n_dst, HC, Kin);
  attn_score<<<cdivl((long)n_src * H, 8), 256, 0, s>>>(w.HS, att_src, w.AS, n_src, H, C);
  attn_score<<<cdivl((long)n_dst * H, 8), 256, 0, s>>>(w.HD, att_dst, w.AD, n_dst, H, C);
  fill_f32<<<cdivl((long)n_dst * H, 256), 256, 0, s>>>(w.AMAX, -INFINITY, n_dst * H);
  fill_f32<<<cdivl((long)n_dst * H, 256), 256, 0, s>>>(w.DENOM, 0.0f, n_dst * H);
  fill_f32<<<cdivl((long)n_dst * HC, 256), 256, 0, s>>>(OUT, 0.0f, n_dst * HC);
  const long Etot = (long)E + nself;
  edge_alpha_max<<<cdivl(Etot * H, 256), 256, 0, s>>>(es, ed, w.AS, w.AD, w.ALPHA, w.AMAX, E, nself, H);
  edge_exp_sum<<<cdivl(Etot * H, 256), 256, 0, s>>>(ed, w.ALPHA, w.AMAX, w.DENOM, E, nself, H);
  edge_scatter4<<<cdivl(Etot * (HC >> 2), 256), 256, 0, s>>>(es, ed, w.ALPHA, w.DENOM, w.HS, OUT,
                                                             E, nself, H, C);
}

extern "C" void kernel_launch(void* const* d_in, const int* in_sizes, int n_in,
                              void* d_out, int out_size, void* d_ws, size_t ws_size,
                              hipStream_t stream) {
  (void)n_in; (void)out_size; (void)ws_size;
  const int NITEM = 10000, NSEQ = 8000, WED = 300, Dm = 128, H = 4, OUTC = 64, HID = 512;
  const int NSELF = 8000;                    // min(NUM_SEQ, NUM_ITEM) self-loops
  const float* x_item = (const float*)d_in[0];
  const float* x_seq  = (const float*)d_in[1];
  const int*   edges  = (const int*)d_in[2];
  const int E = in_sizes[2] / 2;
  const int* e_seq  = edges;                 // edge_index[0]
  const int* e_item = edges + E;             // edge_index[1]
  const float* Wi_w = (const float*)d_in[3];
  const float* Wi_b = (const float*)d_in[4];
  const float* Ws_w = (const float*)d_in[5];
  const float* Ws_b = (const float*)d_in[6];
  // conv param blocks (w_src, w_dst, att_src, att_dst, bias):
  //   c1_si: 7..11, c1_is: 12..16, c2_si: 17..21, c2_is: 22..26

  float* ws = (float*)d_ws;
  size_t off = 0;
  float* h_item  = ws + off; off += (size_t)NITEM * Dm;
  float* h_seq   = ws + off; off += (size_t)NSEQ * Dm;
  float* h_item2 = ws + off; off += (size_t)NITEM * HID;
  float* h_seq2  = ws + off; off += (size_t)NSEQ * HID;
  ConvScratch cs;
  cs.HS    = ws + off; off += (size_t)NITEM * HID;   // max(n_src) x max(HC)
  cs.HD    = ws + off; off += (size_t)NITEM * HID;
  cs.AS    = ws + off; off += (size_t)NITEM * H;
  cs.AD    = ws + off; off += (size_t)NITEM * H;
  cs.AMAX  = ws + off; off += (size_t)NITEM * H;
  cs.DENOM = ws + off; off += (size_t)NITEM * H;
  cs.ALPHA = ws + off; off += (size_t)(E + NSELF) * H;

  float* out_item3 = (float*)d_out;                       // (10000, 64)
  float* out_seq3  = out_item3 + (size_t)NITEM * OUTC;    // (8000, 64)

  // input projections
  launch_gemm(stream, x_item, Wi_w, Wi_b, h_item, NITEM, Dm, WED);
  launch_gemm(stream, x_seq,  Ws_w, Ws_b, h_seq,  NSEQ,  Dm, NITEM);

  // layer 1: seq->item
  run_conv(stream, h_seq, NSEQ, h_item, NITEM, Dm,
           (const float*)d_in[7], (const float*)d_in[8],
           (const float*)d_in[9], (const float*)d_in[10],
           e_seq, e_item, E, NSELF, H, Dm, h_item2, cs);
  bias_act_elu<<<cdivl((long)NITEM * HID, 256), 256, 0, stream>>>(
      h_item2, (const float*)d_in[11], NITEM, HID);

  // layer 1: item->seq
  run_conv(stream, h_item, NITEM, h_seq, NSEQ, Dm,
           (const float*)d_in[12], (const float*)d_in[13],
           (const float*)d_in[14], (const float*)d_in[15],
           e_item, e_seq, E, NSELF, H, Dm, h_seq2, cs);
  bias_act_elu<<<cdivl((long)NSEQ * HID, 256), 256, 0, stream>>>(
      h_seq2, (const float*)d_in[16], NSEQ, HID);

  // layer 2: seq2->item2, softmax rows -> d_out part 1
  run_conv(stream, h_seq2, NSEQ, h_item2, NITEM, HID,
           (const float*)d_in[17], (const float*)d_in[18],
           (const float*)d_in[19], (const float*)d_in[20],
           e_seq, e_item, E, NSELF, 1, OUTC, out_item3, cs);
  bias_softmax64<<<cdivl(NITEM, 8), 256, 0, stream>>>(out_item3, (const float*)d_in[21], NITEM);

  // layer 2: item2->seq2, softmax rows -> d_out part 2
  run_conv(stream, h_item2, NITEM, h_seq2, NSEQ, HID,
           (const float*)d_in[22], (const float*)d_in[23],
           (const float*)d_in[24], (const float*)d_in[25],
           e_item, e_seq, E, NSELF, 1, OUTC, out_seq3, cs);
  bias_softmax64<<<cdivl(NSEQ, 8), 256, 0, stream>>>(out_seq3, (const float*)d_in[26], NSEQ);
}